// GlobalPropagation_84954453115143
// MI455X (gfx1250) — compile-verified
//
#include <hip/hip_runtime.h>
#include <stdint.h>

#define TPB 256
#define WPB 8           // waves per block (wave32 -> 8 waves / 256 threads)
#define MAXSTAGE 128    // LDS-staged edges per row (Poisson(32) max deg ~70)

// ---------------- CDNA5 async-to-LDS support (compile-probed) ----------------
#if defined(__AMDGCN__) && __has_builtin(__builtin_amdgcn_global_load_async_to_lds_b128)
#define USE_ASYNC_LDS 1
#else
#define USE_ASYNC_LDS 0
#endif

#if defined(__AMDGCN__)
typedef int v4i __attribute__((ext_vector_type(4)));
typedef __attribute__((address_space(1))) v4i gv4i;   // global-AS int4
typedef __attribute__((address_space(3))) v4i lv4i;   // LDS-AS int4
#endif

__device__ __forceinline__ void async_copy16(const void* gsrc, const void* ldst) {
#if USE_ASYNC_LDS
    gv4i* g = (gv4i*)(uintptr_t)gsrc;                       // generic global ptr value == AS1 value
    lv4i* l = (lv4i*)(uint32_t)(uintptr_t)ldst;             // low 32 bits == LDS byte offset
    __builtin_amdgcn_global_load_async_to_lds_b128(g, l, 0, 0);
#else
    (void)gsrc; (void)ldst;
#endif
}

__device__ __forceinline__ void wait_async0() {
#if __has_builtin(__builtin_amdgcn_s_wait_asynccnt)
    __builtin_amdgcn_s_wait_asynccnt(0);
#else
    asm volatile("s_wait_asynccnt 0" ::: "memory");
#endif
}

// Non-temporal load for the streamed edge list (setup-only data: keep it out
// of the 192MB L2 so H / CSR / ping-pong stay resident for the 3 SpMM hops).
__device__ __forceinline__ long long nt_load_i64(const long long* p) {
#if __has_builtin(__builtin_nontemporal_load)
    return __builtin_nontemporal_load(p);
#else
    return *p;
#endif
}

// ---------------- setup kernels ----------------
__global__ __launch_bounds__(TPB) void k_zero(unsigned* __restrict__ p, int n) {
    int i = blockIdx.x * TPB + threadIdx.x;
    if (i < n) p[i] = 0u;
}

__global__ __launch_bounds__(TPB) void k_degree(const long long* __restrict__ rows,
                                                unsigned* __restrict__ deg, int E) {
    int e = blockIdx.x * TPB + threadIdx.x;
    if (e < E) atomicAdd(&deg[(int)nt_load_i64(rows + e)], 1u);
}

__global__ __launch_bounds__(TPB) void k_dis(const unsigned* __restrict__ deg,
                                             float* __restrict__ dis, int N) {
    int i = blockIdx.x * TPB + threadIdx.x;
    if (i < N) dis[i] = rsqrtf((float)(deg[i] + 1u));   // +1 = self loop
}

// exclusive scan of ceil4(deg) -> rowptr (pads each row segment to a multiple
// of 4 edges so every segment start is 16B aligned for async b128 staging)
__global__ __launch_bounds__(TPB) void k_scan_block(const unsigned* __restrict__ deg,
                                                    unsigned* __restrict__ rowptr,
                                                    unsigned* __restrict__ bsum, int N) {
    __shared__ unsigned s[TPB];
    int i = blockIdx.x * TPB + threadIdx.x;
    unsigned v = (i < N) ? ((deg[i] + 3u) & ~3u) : 0u;
    s[threadIdx.x] = v;
    __syncthreads();
    for (int o = 1; o < TPB; o <<= 1) {
        unsigned t = (threadIdx.x >= (unsigned)o) ? s[threadIdx.x - o] : 0u;
        __syncthreads();
        s[threadIdx.x] += t;
        __syncthreads();
    }
    if (i < N) rowptr[i] = s[threadIdx.x] - v;          // exclusive
    if (threadIdx.x == TPB - 1) bsum[blockIdx.x] = s[TPB - 1];
}

__global__ void k_scan_sums(unsigned* __restrict__ bsum, int nb) {
    __shared__ unsigned s[1024];
    unsigned v = ((int)threadIdx.x < nb) ? bsum[threadIdx.x] : 0u;
    s[threadIdx.x] = v;
    __syncthreads();
    for (int o = 1; o < 1024; o <<= 1) {
        unsigned t = (threadIdx.x >= (unsigned)o) ? s[threadIdx.x - o] : 0u;
        __syncthreads();
        s[threadIdx.x] += t;
        __syncthreads();
    }
    if ((int)threadIdx.x < nb) bsum[threadIdx.x] = s[threadIdx.x] - v;  // exclusive
}

__global__ __launch_bounds__(TPB) void k_scan_add(unsigned* __restrict__ rowptr,
                                                  const unsigned* __restrict__ bsum, int N) {
    int i = blockIdx.x * TPB + threadIdx.x;
    if (i < N) rowptr[i] += bsum[blockIdx.x];
}

__global__ __launch_bounds__(TPB) void k_fill(const long long* __restrict__ rows,
                                              const long long* __restrict__ cols,
                                              const unsigned* __restrict__ rowptr,
                                              unsigned* __restrict__ rowfill,
                                              const float* __restrict__ dis,
                                              unsigned* __restrict__ colidx,
                                              float* __restrict__ wv, int E) {
    int e = blockIdx.x * TPB + threadIdx.x;
    if (e < E) {
        int r = (int)nt_load_i64(rows + e);
        int c = (int)nt_load_i64(cols + e);
        unsigned pos = rowptr[r] + atomicAdd(&rowfill[r], 1u);
        colidx[pos] = (unsigned)c;
        wv[pos]     = dis[r] * dis[c];
    }
}

// ---------------- SpMM hop: wave-per-row gather, atomic-free ----------------
__global__ __launch_bounds__(TPB) void k_spmm(const float4* __restrict__ Hin,
                                              float4* __restrict__ Hout,
                                              const unsigned* __restrict__ rowptr,
                                              const unsigned* __restrict__ deg,
                                              const float* __restrict__ dis,
                                              const unsigned* __restrict__ colidx,
                                              const float* __restrict__ wv, int N) {
    __shared__ __align__(16) unsigned s_col[WPB][MAXSTAGE];
    __shared__ __align__(16) float    s_w[WPB][MAXSTAGE];
    const int lane = threadIdx.x & 31;
    const int wave = threadIdx.x >> 5;
    const int row  = blockIdx.x * WPB + wave;

    unsigned base = 0, len = 0, stage = 0;
    if (row < N) {
        base  = rowptr[row];
        len   = deg[row];
        stage = len > MAXSTAGE ? MAXSTAGE : len;
    }

#if USE_ASYNC_LDS
    if (row < N) {
        const unsigned nchunk = (stage + 3u) >> 2;     // 16B chunks, <= 32
        if ((unsigned)lane < nchunk) {
            async_copy16(colidx + base + 4 * lane, &s_col[wave][4 * lane]);
            async_copy16(wv     + base + 4 * lane, &s_w[wave][4 * lane]);
        }
        wait_async0();
    }
#else
    if (row < N) {
        for (unsigned k = lane; k < stage; k += 32) {
            s_col[wave][k] = colidx[base + k];
            s_w[wave][k]   = wv[base + k];
        }
    }
#endif
    __syncthreads();

    if (row >= N) return;

    const float di = dis[row];
    const float wself = di * di;                        // self-loop weight
    float4 h = Hin[(size_t)row * 32 + lane];
    float ax = wself * h.x, ay = wself * h.y, az = wself * h.z, aw = wself * h.w;

    #pragma unroll 4
    for (unsigned k = 0; k < stage; ++k) {
        const unsigned c = s_col[wave][k];
        const float wgt  = s_w[wave][k];
        float4 v = Hin[(size_t)c * 32 + lane];          // global_load_b128, L2-resident
        ax += wgt * v.x; ay += wgt * v.y; az += wgt * v.z; aw += wgt * v.w;
    }
    for (unsigned k = stage; k < len; ++k) {            // rare long-row fallback
        const unsigned c = colidx[base + k];
        const float wgt  = wv[base + k];
        float4 v = Hin[(size_t)c * 32 + lane];
        ax += wgt * v.x; ay += wgt * v.y; az += wgt * v.z; aw += wgt * v.w;
    }
    float4 o; o.x = ax; o.y = ay; o.z = az; o.w = aw;
    Hout[(size_t)row * 32 + lane] = o;
}

// ---------------- launcher ----------------
extern "C" void kernel_launch(void* const* d_in, const int* in_sizes, int n_in,
                              void* d_out, int out_size, void* d_ws, size_t ws_size,
                              hipStream_t stream) {
    (void)n_in; (void)out_size; (void)ws_size;
    const float*     H0 = (const float*)d_in[0];
    const long long* el = (const long long*)d_in[1];
    const int N = in_sizes[0] / 128;        // D = 128 (reference constant)
    const int E = in_sizes[1] / 2;
    const long long* rows = el;
    const long long* cols = el + E;

    // workspace carve (256B aligned blocks)
    char* w = (char*)d_ws;
    size_t off = 0;
    auto take = [&](size_t bytes) -> char* {
        char* p = w + off;
        off = (off + bytes + 255) & ~(size_t)255;
        return p;
    };
    unsigned* deg     = (unsigned*)take((size_t)2 * N * 4);  // deg + rowfill contiguous
    unsigned* rowfill = deg + N;
    unsigned* rowptr  = (unsigned*)take((size_t)N * 4);
    unsigned* bsum    = (unsigned*)take((size_t)1024 * 4);
    float*    dis     = (float*)take((size_t)N * 4);
    const size_t cap  = (size_t)E + 4 * (size_t)N;           // padded CSR capacity
    unsigned* colidx  = (unsigned*)take(cap * 4);
    float*    wv      = (float*)take(cap * 4);
    float*    Hmid    = (float*)take((size_t)N * 128 * 4);

    const int gE  = (E + TPB - 1) / TPB;
    const int gN  = (N + TPB - 1) / TPB;
    const int g2N = (2 * N + TPB - 1) / TPB;
    const int gR  = (N + WPB - 1) / WPB;

    k_zero      <<<g2N, TPB, 0, stream>>>(deg, 2 * N);
    k_degree    <<<gE,  TPB, 0, stream>>>(rows, deg, E);
    k_dis       <<<gN,  TPB, 0, stream>>>(deg, dis, N);
    k_scan_block<<<gN,  TPB, 0, stream>>>(deg, rowptr, bsum, N);
    k_scan_sums <<<1,  1024, 0, stream>>>(bsum, gN);         // gN = 391 <= 1024
    k_scan_add  <<<gN,  TPB, 0, stream>>>(rowptr, bsum, N);
    k_fill      <<<gE,  TPB, 0, stream>>>(rows, cols, rowptr, rowfill, dis, colidx, wv, E);

    float* Hout = (float*)d_out;
    k_spmm<<<gR, TPB, 0, stream>>>((const float4*)H0,   (float4*)Hout, rowptr, deg, dis, colidx, wv, N);
    k_spmm<<<gR, TPB, 0, stream>>>((const float4*)Hout, (float4*)Hmid, rowptr, deg, dis, colidx, wv, N);
    k_spmm<<<gR, TPB, 0, stream>>>((const float4*)Hmid, (float4*)Hout, rowptr, deg, dis, colidx, wv, N);
}